// KPBEVEncoder_87239375717064
// MI455X (gfx1250) — compile-verified
//
#include <hip/hip_runtime.h>
#include <math.h>

// ---- constants from the reference ----
#define PCR0f (-51.2f)
#define PCR1f (-51.2f)
#define PCR2f (-5.0f)
#define PCR5f (3.0f)
#define HEIGHTf 8.0f          // PCR5 - PCR2
#define KMAX 8
#define NKPc 15
#define EPSc 1e-5f

typedef float v2f __attribute__((ext_vector_type(2)));
typedef float v8f __attribute__((ext_vector_type(8)));

// -------------------------------------------------------------------------
__global__ void zero_u32_k(unsigned* p, long n) {
  long i = (long)blockIdx.x * blockDim.x + threadIdx.x;
  long stride = (long)gridDim.x * blockDim.x;
  for (; i < n; i += stride) p[i] = 0u;
}

// -------------------------------------------------------------------------
// Dense voxelization: keep first K points per BEV cell (atomic rank).
__global__ void scatter_k(const float* __restrict__ pts, int npts,
                          int* __restrict__ vcount, float* __restrict__ vox,
                          float vs, int G) {
  int i = blockIdx.x * blockDim.x + threadIdx.x;
  if (i >= npts) return;
  const float* p = pts + (size_t)i * 5;
  float x = p[0], y = p[1], z = p[2];
  int ix = (int)floorf((x - PCR0f) / vs);
  int iy = (int)floorf((y - PCR1f) / vs);
  if (ix < 0 || ix >= G || iy < 0 || iy >= G || z < PCR2f || z >= PCR5f) return;
  int vid = iy * G + ix;
  int r = atomicAdd(&vcount[vid], 1);
  if (r < KMAX) {
    float* d = vox + ((size_t)vid * KMAX + r) * 5;
    d[0] = x; d[1] = y; d[2] = z; d[3] = p[3]; d[4] = p[4];
  }
}

// -------------------------------------------------------------------------
// Pass 1: per-block partial sums of masked y and y^2 (for global norm).
__global__ __launch_bounds__(128) void prenorm_k(
    const float* __restrict__ vox, const int* __restrict__ vcount,
    const float* __restrict__ w_pre, const float* __restrict__ b_pre,
    float* __restrict__ partial, int G, float vs) {
  __shared__ float pls[16 * 8 * 5];
  __shared__ float cent[16][3];
  __shared__ float anch[16][3];
  __shared__ int   cntl[16];
  __shared__ float wl[12 * 64];
  __shared__ float psum[16][64];
  __shared__ float psq[16][64];
  int tid = threadIdx.x;
  int v0 = blockIdx.x * 16;
  for (int idx = tid; idx < 16 * 8 * 5; idx += 128)
    pls[idx] = vox[(size_t)v0 * 40 + idx];
  for (int idx = tid; idx < 768; idx += 128) wl[idx] = w_pre[idx];
  if (tid < 16) {
    int vg = v0 + tid;
    int c = vcount[vg]; if (c > KMAX) c = KMAX;
    cntl[tid] = c;
    float sx = 0, sy = 0, sz = 0;
    for (int k = 0; k < c; k++) {
      const float* q = &pls[(tid * 8 + k) * 5];
      sx += q[0]; sy += q[1]; sz += q[2];
    }
    float nf = (float)(c > 0 ? c : 1);
    cent[tid][0] = sx / nf; cent[tid][1] = sy / nf; cent[tid][2] = sz / nf;
    int ix = vg % G, iy = vg / G;
    anch[tid][0] = (float)ix + vs * 0.5f;   // reference quirk: index + meters
    anch[tid][1] = (float)iy + vs * 0.5f;
    anch[tid][2] = HEIGHTf * 0.5f;
  }
  __syncthreads();
  int v = tid >> 3, cg = tid & 7;
  float s8[8], q8[8];
  for (int i = 0; i < 8; i++) { s8[i] = 0.f; q8[i] = 0.f; }
  int cnt = cntl[v];
  for (int k = 0; k < cnt; k++) {
    const float* q = &pls[(v * 8 + k) * 5];
    float x12[12];
    x12[0] = q[3]; x12[1] = q[4];
    x12[2] = q[0] - anch[v][0]; x12[3] = q[1] - anch[v][1]; x12[4] = q[2] - anch[v][2];
    x12[5] = q[0] - cent[v][0]; x12[6] = q[1] - cent[v][1]; x12[7] = q[2] - cent[v][2];
    x12[8] = cent[v][0]; x12[9] = cent[v][1]; x12[10] = cent[v][2];
    x12[11] = (float)cnt;
    for (int cc = 0; cc < 8; cc++) {
      int c = cg * 8 + cc;
      float y = b_pre[c];
      #pragma unroll
      for (int i = 0; i < 12; i++) y += x12[i] * wl[i * 64 + c];
      s8[cc] += y; q8[cc] += y * y;
    }
  }
  for (int cc = 0; cc < 8; cc++) { psum[v][cg * 8 + cc] = s8[cc]; psq[v][cg * 8 + cc] = q8[cc]; }
  __syncthreads();
  if (tid < 64) {
    float a = 0, b = 0;
    for (int vv = 0; vv < 16; vv++) { a += psum[vv][tid]; b += psq[vv][tid]; }
    partial[(size_t)blockIdx.x * 130 + tid] = a;
    partial[(size_t)blockIdx.x * 130 + 64 + tid] = b;
  }
  if (tid == 64) {
    int cs = 0;
    for (int vv = 0; vv < 16; vv++) cs += cntl[vv];
    partial[(size_t)blockIdx.x * 130 + 128] = (float)cs;
  }
}

__global__ void reduce_pre_k(const float* __restrict__ partial, int nblocks,
                             float* __restrict__ params) {
  int c = threadIdx.x;  // 64
  float s = 0, q = 0, cnt = 0;
  for (int b = 0; b < nblocks; b++) {
    s   += partial[(size_t)b * 130 + c];
    q   += partial[(size_t)b * 130 + 64 + c];
    cnt += partial[(size_t)b * 130 + 128];
  }
  float cf = cnt > 1.f ? cnt : 1.f;
  float m  = s / cf;
  float var = q / cf - m * m;
  if (var < 0.f) var = 0.f;
  params[c] = m;
  params[64 + c] = rsqrtf(var + EPSc);
}

// -------------------------------------------------------------------------
// Main fused kernel: 4 waves, 16 voxels per block.
//   S[16,960] = h^T y  (materialized in LDS in 4 chunks of 240, cooperatively)
//   out[16,64] = S @ KPW[960,64]          (240 x v_wmma_f32_16x16x4_f32)
//   z[16,64]   = out @ Wpost[64,64] + b   ( 16 x v_wmma_f32_16x16x4_f32)
// LDS overlays (floats):
//   [0..8192)      ybuf      (also z staging at the end)
//   [8192..10240)  hbuf
//   [10240..14080) S chunk / phase-A staging (pts,cnt,cent,anch,wl) / obuf
#define SMEM_F 14080
#define OFF_Y  0
#define OFF_H  8192
#define OFF_S  10240
#define OFF_PLS  10240           // 640
#define OFF_CNT  10880           // 16 (as int)
#define OFF_CENT 10896           // 48
#define OFF_ANCH 10944           // 48
#define OFF_WL   10992           // 768  (ends 11760)

__global__ __launch_bounds__(128) void main_k(
    const float* __restrict__ vox, const int* __restrict__ vcount,
    const float* __restrict__ w_pre, const float* __restrict__ b_pre,
    const float* __restrict__ g_pre, const float* __restrict__ beta_pre,
    const float* __restrict__ kp, const float* __restrict__ kpw,
    const float* __restrict__ w_post, const float* __restrict__ b_post,
    const float* __restrict__ params, float* __restrict__ zg,
    float* __restrict__ partial_post, int G, float vs) {
  __shared__ float smem[SMEM_F];
  float* ybuf = smem + OFF_Y;
  float* hbuf = smem + OFF_H;
  float* sbuf = smem + OFF_S;
  float* pls  = smem + OFF_PLS;
  int*   cntl = (int*)(smem + OFF_CNT);
  float* cent = smem + OFF_CENT;
  float* anch = smem + OFF_ANCH;
  float* wl   = smem + OFF_WL;

  int tid = threadIdx.x;
  int v0 = blockIdx.x * 16;

  for (int idx = tid; idx < 16 * 8 * 5; idx += 128)
    pls[idx] = vox[(size_t)v0 * 40 + idx];
  for (int idx = tid; idx < 768; idx += 128) wl[idx] = w_pre[idx];
  if (tid < 16) {
    int vg = v0 + tid;
    int c = vcount[vg]; if (c > KMAX) c = KMAX;
    cntl[tid] = c;
    float sx = 0, sy = 0, sz = 0;
    for (int k = 0; k < c; k++) {
      const float* q = &pls[(tid * 8 + k) * 5];
      sx += q[0]; sy += q[1]; sz += q[2];
    }
    float nf = (float)(c > 0 ? c : 1);
    cent[tid * 3 + 0] = sx / nf; cent[tid * 3 + 1] = sy / nf; cent[tid * 3 + 2] = sz / nf;
    int ix = vg % G, iy = vg / G;
    anch[tid * 3 + 0] = (float)ix + vs * 0.5f;
    anch[tid * 3 + 1] = (float)iy + vs * 0.5f;
    anch[tid * 3 + 2] = HEIGHTf * 0.5f;
  }
  __syncthreads();

  // ---- y (pre-MLP + global norm + relu + mask) and h into LDS ----
  {
    int v = tid >> 3, k = tid & 7;
    int cnt = cntl[v];
    float maskv = (k < cnt) ? 1.f : 0.f;
    const float* q = &pls[(v * 8 + k) * 5];
    float x12[12];
    x12[0] = q[3]; x12[1] = q[4];
    x12[2] = q[0] - anch[v * 3 + 0]; x12[3] = q[1] - anch[v * 3 + 1]; x12[4] = q[2] - anch[v * 3 + 2];
    x12[5] = q[0] - cent[v * 3 + 0]; x12[6] = q[1] - cent[v * 3 + 1]; x12[7] = q[2] - cent[v * 3 + 2];
    x12[8] = cent[v * 3 + 0]; x12[9] = cent[v * 3 + 1]; x12[10] = cent[v * 3 + 2];
    x12[11] = (float)cnt;
    for (int c = 0; c < 64; c++) {
      float y = b_pre[c];
      #pragma unroll
      for (int i = 0; i < 12; i++) y += x12[i] * wl[i * 64 + c];
      y = (y - params[c]) * params[64 + c] * g_pre[c] + beta_pre[c];
      y = fmaxf(y, 0.f) * maskv;
      ybuf[(v * 8 + k) * 64 + c] = y;
    }
    float dx = q[0] - anch[v * 3 + 0];
    float dy = q[1] - anch[v * 3 + 1];
    float dz = q[2] - anch[v * 3 + 2];
    for (int j = 0; j < NKPc; j++) {
      float ex = dx - kp[j * 3 + 0];
      float ey = dy - kp[j * 3 + 1];
      float ez = dz - kp[j * 3 + 2];
      float dist = sqrtf(ex * ex + ey * ey + ez * ez + 1e-12f);
      hbuf[(v * 8 + k) * 16 + j] = fmaxf(1.f - dist, 0.f) * maskv;  // SIGMA = 1
    }
    hbuf[(v * 8 + k) * 16 + 15] = 0.f;
  }
  __syncthreads();   // pls/wl dead from here; sbuf overlays them

  // ---- GEMM1: wave w owns 16x16 tile (voxels 0..15 x channels 16w..16w+15)
  int lane = tid & 31;
  int w    = tid >> 5;
  int m    = lane & 15;              // A row (voxel) / D column-within-tile
  int kb   = (lane >> 4) << 1;       // K sub-pair owned by this half-wave
  int dcol = w * 16 + m;             // output channel column
  v8f acc = {0.f, 0.f, 0.f, 0.f, 0.f, 0.f, 0.f, 0.f};

  for (int ch = 0; ch < 4; ch++) {
    if (ch) __syncthreads();         // previous chunk fully consumed
    // cooperative fill of S chunk: S[mm][t] for kappa = ch*240 + t
    for (int idx = tid; idx < 16 * 240; idx += 128) {
      int mm = idx / 240;
      int t  = idx - mm * 240;
      int kap = ch * 240 + t;
      int j = kap >> 6;
      int c = kap & 63;
      const float* hb = &hbuf[mm * 128 + j];
      const float* yb = &ybuf[mm * 512 + c];
      float s = 0.f;
      #pragma unroll
      for (int k = 0; k < 8; k++) s += hb[k * 16] * yb[k * 64];
      sbuf[mm * 240 + t] = s;
    }
    __syncthreads();
    for (int kk = 0; kk < 60; kk++) {
      int kl = kk * 4 + kb;          // local kappa within chunk (even)
      int kg = ch * 240 + kl;        // global kappa
      v2f A = {sbuf[m * 240 + kl], sbuf[m * 240 + kl + 1]};
      v2f B = {kpw[(size_t)kg * 64 + dcol], kpw[(size_t)(kg + 1) * 64 + dcol]};
      acc = __builtin_amdgcn_wmma_f32_16x16x4_f32(false, A, false, B,
                                                  (short)0, acc, false, false);
    }
  }
  __syncthreads();                   // S region dead -> reuse as obuf
  float* obuf = sbuf;                // 16*64
  {
    int rowh = (lane >> 4) * 8;      // C/D layout: vgpr r -> M = r (+8 hi half)
    #pragma unroll
    for (int r = 0; r < 8; r++) obuf[(rowh + r) * 64 + dcol] = acc[r];
  }
  __syncthreads();

  // ---- GEMM2: z = out @ w_post + b_post ----
  float bp = b_post[dcol];
  v8f zacc = {bp, bp, bp, bp, bp, bp, bp, bp};
  for (int kk = 0; kk < 16; kk++) {
    int kap = kk * 4 + kb;
    v2f A = {obuf[m * 64 + kap], obuf[m * 64 + kap + 1]};
    v2f B = {w_post[kap * 64 + dcol], w_post[(kap + 1) * 64 + dcol]};
    zacc = __builtin_amdgcn_wmma_f32_16x16x4_f32(false, A, false, B,
                                                 (short)0, zacc, false, false);
  }
  __syncthreads();                   // ybuf dead -> reuse as z staging
  {
    int rowh = (lane >> 4) * 8;
    #pragma unroll
    for (int r = 0; r < 8; r++) ybuf[(rowh + r) * 64 + dcol] = zacc[r];
  }
  __syncthreads();

  if (tid < 64) {                    // per-block partials for post batch-norm
    float s = 0, q = 0;
    for (int row = 0; row < 16; row++) {
      float zv = ybuf[row * 64 + tid];
      s += zv; q += zv * zv;
    }
    partial_post[(size_t)blockIdx.x * 128 + tid] = s;
    partial_post[(size_t)blockIdx.x * 128 + 64 + tid] = q;
  }
  for (int idx = tid; idx < 1024; idx += 128) {
    int row = idx >> 6, c = idx & 63;
    zg[(size_t)(v0 + row) * 64 + c] = ybuf[idx];
  }
}

__global__ void reduce_post_k(const float* __restrict__ partial, int nblocks,
                              int V, float* __restrict__ params) {
  int c = threadIdx.x;  // 64
  float s = 0, q = 0;
  for (int b = 0; b < nblocks; b++) {
    s += partial[(size_t)b * 128 + c];
    q += partial[(size_t)b * 128 + 64 + c];
  }
  float m = s / (float)V;
  float var = q / (float)V - m * m;
  if (var < 0.f) var = 0.f;
  params[128 + c] = m;
  params[192 + c] = rsqrtf(var + EPSc);
}

__global__ void writeout_k(const float* __restrict__ zg,
                           const float* __restrict__ params,
                           const float* __restrict__ g_post,
                           const float* __restrict__ beta_post,
                           float* __restrict__ out, int V) {
  int i = blockIdx.x * blockDim.x + threadIdx.x;
  if (i >= V * 64) return;
  int v = i >> 6, c = i & 63;
  float z = zg[(size_t)v * 64 + c];
  z = (z - params[128 + c]) * params[192 + c] * g_post[c] + beta_post[c];
  z = fmaxf(z, 0.f);
  out[(size_t)c * V + v] = z;      // [64, G, G] channel-major
}

// -------------------------------------------------------------------------
extern "C" void kernel_launch(void* const* d_in, const int* in_sizes, int n_in,
                              void* d_out, int out_size, void* d_ws, size_t ws_size,
                              hipStream_t stream) {
  const float* points    = (const float*)d_in[0];
  const float* kp        = (const float*)d_in[1];
  const float* w_pre     = (const float*)d_in[2];
  const float* b_pre     = (const float*)d_in[3];
  const float* g_pre     = (const float*)d_in[4];
  const float* beta_pre  = (const float*)d_in[5];
  const float* kpw       = (const float*)d_in[6];
  const float* w_post    = (const float*)d_in[7];
  const float* b_post    = (const float*)d_in[8];
  const float* g_post    = (const float*)d_in[9];
  const float* beta_post = (const float*)d_in[10];
  int Npts = in_sizes[0] / 5;

  static const int   GS[4]  = {256, 128, 64, 32};
  static const float VSS[4] = {0.4f, 0.8f, 1.6f, 3.2f};
  static const int   VOFF[4] = {0, 65536, 81920, 86016};
  const int VT = 87040;

  // ---- workspace layout ----
  char*  ws           = (char*)d_ws;
  int*   npts_all     = (int*)ws;                              // VT ints
  float* vox_all      = (float*)(ws + (size_t)VT * 4);         // VT*40 f
  float* zbuf_all     = vox_all + (size_t)VT * 40;             // VT*64 f
  float* partial_pre  = zbuf_all + (size_t)VT * 64;            // 4096*130 f
  float* partial_post = partial_pre + (size_t)4096 * 130;      // 4096*128 f
  float* params       = partial_post + (size_t)4096 * 128;     // 256 f

  // zero voxel counts + voxel payloads (contiguous region) once per call
  long zn = (long)VT + (long)VT * 40;
  zero_u32_k<<<1024, 256, 0, stream>>>((unsigned*)ws, zn);

  float* outp = (float*)d_out;
  size_t ooff = 0;
  for (int s = 0; s < 4; s++) {
    int G = GS[s]; float vs = VSS[s]; int V = G * G;
    int*   vc = npts_all + VOFF[s];
    float* vx = vox_all + (size_t)VOFF[s] * 40;
    float* zg = zbuf_all + (size_t)VOFF[s] * 64;
    int nb = V / 16;

    scatter_k<<<(Npts + 255) / 256, 256, 0, stream>>>(points, Npts, vc, vx, vs, G);
    prenorm_k<<<nb, 128, 0, stream>>>(vx, vc, w_pre, b_pre, partial_pre, G, vs);
    reduce_pre_k<<<1, 64, 0, stream>>>(partial_pre, nb, params);
    main_k<<<nb, 128, 0, stream>>>(vx, vc, w_pre, b_pre, g_pre, beta_pre,
                                   kp, kpw, w_post, b_post, params, zg,
                                   partial_post, G, vs);
    reduce_post_k<<<1, 64, 0, stream>>>(partial_post, nb, V, params);
    writeout_k<<<(V * 64 + 255) / 256, 256, 0, stream>>>(zg, params, g_post,
                                                         beta_post, outp + ooff, V);
    ooff += (size_t)64 * V;
  }
}